// Encoder_22359599743333
// MI455X (gfx1250) — compile-verified
//
#include <hip/hip_runtime.h>
#include <math.h>

// Elman RNN (T=512, B=64, V=50257, E=300, H=512) on gfx1250.
// - Input projection: one big parallel f32 WMMA GEMM (M=32768, N=512, K=300),
//   written directly into the d_out[t] slabs.
// - Recurrence: ONE persistent kernel (16 blocks) iterating all 512 steps,
//   W_hh staged in LDS once, agent-scope atomic barrier between steps.

typedef __attribute__((ext_vector_type(2))) float v2f;
typedef __attribute__((ext_vector_type(4))) float v4f;
typedef __attribute__((ext_vector_type(8))) float v8f;

#define T_ 512
#define B_ 64
#define E_ 300
#define H_ 512
#define NBLK 16   // persistent blocks: 4 M-tiles x 32 N-tiles = 128 tiles / 8 waves

__device__ __forceinline__ v2f ld2(const float* p) { return *(const v2f*)p; }
__device__ __forceinline__ v4f ld4(const float* p) { return *(const v4f*)p; }
__device__ __forceinline__ v2f lo2(v4f v) { return __builtin_shufflevector(v, v, 0, 1); }
__device__ __forceinline__ v2f hi2(v4f v) { return __builtin_shufflevector(v, v, 2, 3); }

__device__ __forceinline__ v8f wmma_f32(v2f a, v2f b, v8f c) {
    return __builtin_amdgcn_wmma_f32_16x16x4_f32(
        /*neg_a=*/false, a, /*neg_b=*/false, b,
        /*c_mod=*/(short)0, c, /*reuse_a=*/false, /*reuse_b=*/false);
}

// ---------------------------------------------------------------------------
// Kernel 1: xp[row, h] = sum_e emb[src[row], e] * W_ih[h, e] + b_ih[h] + b_hh[h]
// One 16x16 tile per wave. K paired: one b128 load feeds two WMMAs
// (half-wave 0 supplies k0..k0+3, half-wave 1 supplies k0+4..k0+7; the
// K-permutation is consistent between A and B, so the dot product is exact).
// ---------------------------------------------------------------------------
__global__ __launch_bounds__(256)
void rnn_xp_kernel(const int* __restrict__ src, const float* __restrict__ emb,
                   const float* __restrict__ W_ih, const float* __restrict__ b_ih,
                   const float* __restrict__ b_hh, float* __restrict__ xp)
{
    const int lane   = threadIdx.x & 31;
    const int wave   = blockIdx.x * (blockDim.x >> 5) + (threadIdx.x >> 5);
    const int tile_m = wave >> 5;      // 0..2047
    const int tile_n = wave & 31;      // 0..31
    const int m0 = tile_m * 16, n0 = tile_n * 16;
    const int half = lane >> 4, l15 = lane & 15;

    const int    tok  = src[m0 + l15];
    const float* arow = emb  + (size_t)tok * E_;
    const float* brow = W_ih + (size_t)(n0 + l15) * E_;

    v8f c = {0.f, 0.f, 0.f, 0.f, 0.f, 0.f, 0.f, 0.f};
    for (int k0 = 0; k0 < 296; k0 += 8) {          // 37 paired iterations
        v4f a = ld4(arow + k0 + half * 4);
        v4f b = ld4(brow + k0 + half * 4);
        c = wmma_f32(lo2(a), lo2(b), c);
        c = wmma_f32(hi2(a), hi2(b), c);
    }
    {                                              // K tail: 296..299
        v2f a = ld2(arow + 296 + half * 2);
        v2f b = ld2(brow + 296 + half * 2);
        c = wmma_f32(a, b, c);
    }

    const float bias = b_ih[n0 + l15] + b_hh[n0 + l15];
    float* out = xp + (size_t)m0 * H_ + n0 + l15;
    #pragma unroll
    for (int r = 0; r < 8; ++r)
        out[(size_t)(r + half * 8) * H_] = c[r] + bias;
}

// ---------------------------------------------------------------------------
// Kernel 2: persistent scan over all T timesteps.
// Block b owns tile_n in {2b, 2b+1} and tile_m in {0..3} (8 waves).
// W_hh rows [32b, 32b+32) staged in 64 KB LDS once; per step the inner K loop
// reads A (h_{t-1}) from global/L2 and B from LDS, then tanh-epilogues in
// place over the xp slab. Steps are separated by an agent-scope barrier.
// ---------------------------------------------------------------------------
__global__ __launch_bounds__(256)
void rnn_scan_kernel(const float* __restrict__ W_hh, float* __restrict__ out,
                     unsigned int* __restrict__ bar)
{
    __shared__ float w_lds[32 * H_];               // 64 KB

    const int tid  = threadIdx.x;
    const int lane = tid & 31;
    const int ww   = tid >> 5;                     // wave in block: 0..7
    const int blk  = blockIdx.x;                   // 0..15
    const int tile_n = 2 * blk + (ww >> 2);
    const int tile_m = ww & 3;
    const int m0 = tile_m * 16, n0 = tile_n * 16;
    const int half = lane >> 4, l15 = lane & 15;

    // One-time stage of this block's 32 W_hh rows into LDS (b128 copies).
    {
        const v4f* s4 = (const v4f*)(W_hh + (size_t)blk * 32 * H_);
        v4f*       d4 = (v4f*)w_lds;
        for (int i = tid; i < 32 * H_ / 4; i += 256) d4[i] = s4[i];
    }
    __syncthreads();

    const float* brow = w_lds + (size_t)((ww >> 2) * 16 + l15) * H_;

    for (int t = 0; t < T_; ++t) {
        v8f c = {0.f, 0.f, 0.f, 0.f, 0.f, 0.f, 0.f, 0.f};
        if (t > 0) {                               // wave-uniform: EXEC stays all-1s
            const float* arow =
                out + (size_t)(t - 1) * B_ * H_ + (size_t)(m0 + l15) * H_;
            for (int k0 = 0; k0 < H_; k0 += 8) {   // 64 paired iterations
                v4f a = ld4(arow + k0 + half * 4); // global (L2-resident slab)
                v4f b = ld4(brow + k0 + half * 4); // LDS
                c = wmma_f32(lo2(a), lo2(b), c);
                c = wmma_f32(hi2(a), hi2(b), c);
            }
        }

        // Epilogue: h_t = tanh(xp_t + h_{t-1} W_hh^T), in place over the slab.
        float* p = out + (size_t)t * B_ * H_ + (size_t)m0 * H_ + n0 + l15;
        #pragma unroll
        for (int r = 0; r < 8; ++r) {
            const size_t off = (size_t)(r + half * 8) * H_;
            p[off] = tanhf(p[off] + c[r]);
        }

        // Device-wide step barrier: make h_t visible, then rendezvous.
        __threadfence();                           // each thread drains its stores
        __syncthreads();
        if (tid == 0) {
            __hip_atomic_fetch_add(bar, 1u, __ATOMIC_RELEASE,
                                   __HIP_MEMORY_SCOPE_AGENT);
            const unsigned int tgt = (unsigned int)NBLK * (unsigned int)(t + 1);
            while (__hip_atomic_load(bar, __ATOMIC_ACQUIRE,
                                     __HIP_MEMORY_SCOPE_AGENT) < tgt) {
                __builtin_amdgcn_s_sleep(2);
            }
        }
        __syncthreads();
    }
}

// ---------------------------------------------------------------------------
extern "C" void kernel_launch(void* const* d_in, const int* in_sizes, int n_in,
                              void* d_out, int out_size, void* d_ws, size_t ws_size,
                              hipStream_t stream) {
    (void)in_sizes; (void)n_in; (void)out_size; (void)ws_size;

    const int*   src  = (const int*)  d_in[0];
    const float* emb  = (const float*)d_in[1];
    const float* W_ih = (const float*)d_in[2];
    const float* W_hh = (const float*)d_in[3];
    const float* b_ih = (const float*)d_in[4];
    const float* b_hh = (const float*)d_in[5];
    float* out = (float*)d_out;
    unsigned int* bar = (unsigned int*)d_ws;

    // Barrier counter must start at 0 every call (d_ws is not re-poisoned).
    hipMemsetAsync(bar, 0, sizeof(unsigned int), stream);

    // Phase 1: xp for all timesteps into the output slabs (fully parallel).
    rnn_xp_kernel<<<8192, 256, 0, stream>>>(src, emb, W_ih, b_ih, b_hh, out);

    // Phase 2: one persistent kernel scans all 512 timesteps.
    rnn_scan_kernel<<<NBLK, 256, 0, stream>>>(W_hh, out, bar);

    // h_last[None] = outs[T-1], appended after the [T,B,H] block.
    hipMemcpyAsync(out + (size_t)T_ * B_ * H_,
                   out + (size_t)(T_ - 1) * B_ * H_,
                   (size_t)B_ * H_ * sizeof(float),
                   hipMemcpyDeviceToDevice, stream);
}